// Qwen3NextGatedDeltaNet_41343355191681
// MI455X (gfx1250) — compile-verified
//
#include <hip/hip_runtime.h>
#include <math.h>

// ---------------------------------------------------------------------------
// Problem dimensions (fixed by the reference)
// ---------------------------------------------------------------------------
#define LSEQ     4096
#define EDIM     2048
#define HV       32
#define HK       16
#define DK       128
#define DV       128
#define CHUNK    64
#define NCHUNK   64
#define KEY_DIM  2048      // HK*DK
#define VALUE_DIM 4096     // HV*DV
#define QKVZ_N   12288     // 2*KEY_DIM + 2*VALUE_DIM
#define CONV_DIM 8192      // 2*KEY_DIM + VALUE_DIM

typedef float v2f __attribute__((ext_vector_type(2)));
typedef float v8f __attribute__((ext_vector_type(8)));

// D = A(16x4) * B(4x16) + C, fp32 WMMA (CDNA5)
__device__ __forceinline__ v8f wmma4(v2f a, v2f b, v8f c) {
  return __builtin_amdgcn_wmma_f32_16x16x4_f32(
      /*neg_a=*/false, a, /*neg_b=*/false, b,
      /*c_mod=*/(short)0, c, /*reuse_a=*/false, /*reuse_b=*/false);
}

__device__ __forceinline__ float silu_f(float x) {
  return x / (1.f + expf(-x));
}

// ---------------------------------------------------------------------------
// fp32 WMMA GEMM:  D[M,N] = A[M,K] @ W[K,N]   (M%128==0, N%64==0, K%32==0)
// 256 threads (8 waves), 128x64 block tile; wave = 2x2 grid of 16x16 tiles.
// A staged row-major [m][k]; W staged TRANSPOSED [n][k] so every WMMA
// fragment is one aligned b64 LDS load.
// ---------------------------------------------------------------------------
#define GT_K 32
#define LDT  (GT_K + 2)   // 34: even stride -> 8B-aligned rows, conflict-free
__global__ __launch_bounds__(256) void gemm_f32_wmma(
    const float* __restrict__ A, const float* __restrict__ W,
    float* __restrict__ D, int M, int N, int Kd) {
  __shared__ float As[128][LDT];
  __shared__ float Wt[64][LDT];
  const int tid  = threadIdx.x;
  const int wave = tid >> 5, lane = tid & 31;
  const int lr = lane & 15, lh = lane >> 4;
  const int mBase = blockIdx.x * 128;
  const int nBase = blockIdx.y * 64;
  const int wm = (wave & 3) * 32;   // wave's 32-row stripe (2 M tiles)
  const int wn = (wave >> 2) * 32;  // wave's 32-col stripe (2 N tiles)
  v8f acc00 = {}, acc01 = {}, acc10 = {}, acc11 = {};

  const int ra = tid >> 1, ca = (tid & 1) * 16;  // A: 128x32, 16 floats/thread
  const int rw = tid >> 3, cw = (tid & 7) * 8;   // W: 32x64,   8 floats/thread

  for (int k0 = 0; k0 < Kd; k0 += GT_K) {
    const float* srcA = A + (size_t)(mBase + ra) * Kd + k0 + ca;
    #pragma unroll
    for (int j = 0; j < 16; ++j) As[ra][ca + j] = srcA[j];
    const float* srcW = W + (size_t)(k0 + rw) * N + nBase + cw;
    #pragma unroll
    for (int j = 0; j < 8; ++j) Wt[cw + j][rw] = srcW[j];   // transpose
    if (k0 + GT_K < Kd) {
      __builtin_prefetch(srcA + GT_K, 0, 1);
      __builtin_prefetch(srcW + (size_t)GT_K * N, 0, 1);
    }
    __syncthreads();
    #pragma unroll
    for (int kk = 0; kk < GT_K; kk += 4) {
      const int ko = kk + 2 * lh;
      v2f a0 = *(const v2f*)&As[wm + lr][ko];
      v2f a1 = *(const v2f*)&As[wm + 16 + lr][ko];
      v2f b0 = *(const v2f*)&Wt[wn + lr][ko];
      v2f b1 = *(const v2f*)&Wt[wn + 16 + lr][ko];
      acc00 = wmma4(a0, b0, acc00);
      acc01 = wmma4(a0, b1, acc01);
      acc10 = wmma4(a1, b0, acc10);
      acc11 = wmma4(a1, b1, acc11);
    }
    __syncthreads();
  }
  const int m0 = mBase + wm, n0 = nBase + wn;
  #pragma unroll
  for (int v = 0; v < 8; ++v) {
    const int r0 = m0 + v + 8 * lh;
    D[(size_t)r0 * N + n0 + lr]             = acc00[v];
    D[(size_t)r0 * N + n0 + 16 + lr]        = acc01[v];
    D[(size_t)(r0 + 16) * N + n0 + lr]      = acc10[v];
    D[(size_t)(r0 + 16) * N + n0 + 16 + lr] = acc11[v];
  }
}

// ---------------------------------------------------------------------------
// beta = sigmoid(b), g = -exp(A_log)*softplus(a+dt_bias), stored head-major
// ---------------------------------------------------------------------------
__global__ void gb_kernel(const float* __restrict__ ba,
                          const float* __restrict__ A_log,
                          const float* __restrict__ dt_bias,
                          float* __restrict__ g_h, float* __restrict__ beta_h) {
  int idx = blockIdx.x * blockDim.x + threadIdx.x;
  if (idx >= LSEQ * HV) return;
  int l = idx / HV, h = idx % HV;
  float b = ba[l * 64 + h];
  float a = ba[l * 64 + 32 + h];
  float x = a + dt_bias[h];
  float sp = (x > 20.f) ? x : log1pf(expf(x));
  g_h[h * LSEQ + l]    = -expf(A_log[h]) * sp;
  beta_h[h * LSEQ + l] = 1.f / (1.f + expf(-b));
}

// ---------------------------------------------------------------------------
// Fused causal conv(K=4)+SiLU, per-head L2-norm; writes head-major qn/kn/vb.
// ---------------------------------------------------------------------------
__global__ __launch_bounds__(128) void prep_kernel(
    const float* __restrict__ qkvz, const float* __restrict__ conv_w,
    const float* __restrict__ conv_b, const float* __restrict__ beta_h,
    float* __restrict__ qn, float* __restrict__ kn, float* __restrict__ vb) {
  const int h = blockIdx.x, l = blockIdx.y, d = threadIdx.x;
  const int kh = h >> 1;
  const int cq = kh * DK + d;
  const int ck = KEY_DIM + kh * DK + d;
  const int cv = 2 * KEY_DIM + h * DV + d;
  float sq = conv_b[cq], sk = conv_b[ck], sv = conv_b[cv];
  #pragma unroll
  for (int j = 0; j < 4; ++j) {
    int li = l - 3 + j;
    if (li >= 0) {
      const float* row = qkvz + (size_t)li * QKVZ_N;
      sq += row[cq] * conv_w[j * CONV_DIM + cq];
      sk += row[ck] * conv_w[j * CONV_DIM + ck];
      sv += row[cv] * conv_w[j * CONV_DIM + cv];
    }
  }
  sq = silu_f(sq); sk = silu_f(sk); sv = silu_f(sv);

  __shared__ float red[128];
  red[d] = sq * sq; __syncthreads();
  for (int s = 64; s > 0; s >>= 1) { if (d < s) red[d] += red[d + s]; __syncthreads(); }
  const float qinv = rsqrtf(red[0] + 1e-6f) * 0.08838834764831845f;  // Dk^-0.5
  __syncthreads();
  red[d] = sk * sk; __syncthreads();
  for (int s = 64; s > 0; s >>= 1) { if (d < s) red[d] += red[d + s]; __syncthreads(); }
  const float kinv = rsqrtf(red[0] + 1e-6f);

  const size_t o = ((size_t)h * LSEQ + l) * DK + d;
  qn[o] = sq * qinv;
  kn[o] = sk * kinv;
  vb[o] = sv * beta_h[h * LSEQ + l];
}

// ---------------------------------------------------------------------------
// Intra-chunk kernel (one block per (chunk n, head h)), 256 threads / 8 waves.
// ---------------------------------------------------------------------------
#define LDK (DK + 2)      // 130
#define LDC (CHUNK + 2)   // 66
__global__ __launch_bounds__(256) void chunk_kernel(
    const float* __restrict__ kn, const float* __restrict__ vb,
    const float* __restrict__ g_h, const float* __restrict__ beta_h,
    float* __restrict__ v_intra, float* __restrict__ kcd) {
  __shared__ float kns[CHUNK][LDK];   // k chunk, row-major [s][d]
  __shared__ float vbsT[DV][LDC];     // v*beta (later k*beta*e^g) TRANSPOSED [d][s]
  __shared__ float attn[CHUNK][LDC];
  __shared__ float graw[CHUNK], gcum[CHUNK], betas[CHUNK], rowc[CHUNK];

  const int n = blockIdx.x, h = blockIdx.y;
  const int tid = threadIdx.x, wave = tid >> 5, lane = tid & 31;
  const int lr = lane & 15, lh = lane >> 4;
  const size_t base = ((size_t)h * LSEQ + n * CHUNK) * DK;

  {
    const int r = tid >> 2, c0 = (tid & 3) * 32;
    for (int j = 0; j < 32; ++j) kns[r][c0 + j] = kn[base + (size_t)r * DK + c0 + j];
    for (int j = 0; j < 32; ++j) vbsT[c0 + j][r] = vb[base + (size_t)r * DK + c0 + j];
  }
  if (tid < CHUNK) {
    graw[tid]  = g_h[h * LSEQ + n * CHUNK + tid];
    betas[tid] = beta_h[h * LSEQ + n * CHUNK + tid];
  }
  __syncthreads();
  if (tid == 0) {
    float s = 0.f;
    for (int i = 0; i < CHUNK; ++i) { s += graw[i]; gcum[i] = s; }
  }
  __syncthreads();

  // attn = -(k*beta @ k^T) * exp(gcum_s - gcum_t), t < s only
  {
    const int t2 = wave * 2;
    for (int tix = t2; tix < t2 + 2; ++tix) {
      const int tm = tix >> 2, tn = tix & 3;
      const int am = tm * 16 + lr, bn = tn * 16 + lr;
      const float bscale = betas[am];
      v8f c = {};
      #pragma unroll 8
      for (int kk = 0; kk < DK; kk += 4) {
        const int ko = kk + 2 * lh;
        v2f a = *(const v2f*)&kns[am][ko];
        a *= bscale;
        v2f b = *(const v2f*)&kns[bn][ko];   // (k^T)[k][n] = kns[n][k]
        c = wmma4(a, b, c);
      }
      #pragma unroll
      for (int v = 0; v < 8; ++v) {
        const int s_ = tm * 16 + v + 8 * lh;
        const int t_ = tn * 16 + lr;
        attn[s_][t_] = (t_ < s_) ? -c[v] * expf(gcum[s_] - gcum[t_]) : 0.f;
      }
    }
  }
  __syncthreads();

  // sequential forward substitution
  for (int i = 1; i < CHUNK; ++i) {
    if (tid < i) rowc[tid] = attn[i][tid];
    __syncthreads();
    if (tid < i) {
      float s = rowc[tid];
      for (int t = 0; t < i; ++t) s += rowc[t] * attn[t][tid];
      attn[i][tid] = s;
    }
    __syncthreads();
  }
  if (tid < CHUNK) attn[tid][tid] = 1.f;
  __syncthreads();

  // v_intra = attn @ (v*beta)   (4 M-tiles x 8 N-tiles, 4 tiles/wave)
  const int tm = wave >> 1, tnb = (wave & 1) * 4;
  const int am = tm * 16 + lr;
  for (int tn = tnb; tn < tnb + 4; ++tn) {
    const int bn = tn * 16 + lr;
    v8f c = {};
    #pragma unroll 8
    for (int kk = 0; kk < CHUNK; kk += 4) {
      const int ko = kk + 2 * lh;
      v2f a = *(const v2f*)&attn[am][ko];
      v2f b = *(const v2f*)&vbsT[bn][ko];
      c = wmma4(a, b, c);
    }
    #pragma unroll
    for (int v = 0; v < 8; ++v)
      v_intra[base + (size_t)(tm * 16 + v + 8 * lh) * DV + bn] = c[v];
  }
  __syncthreads();

  {  // rebuild transposed buffer as kbe = k*beta*exp(g_raw)
    const int r = tid >> 2, c0 = (tid & 3) * 32;
    const float sc = betas[r] * expf(graw[r]);
    for (int j = 0; j < 32; ++j) vbsT[c0 + j][r] = kns[r][c0 + j] * sc;
  }
  __syncthreads();

  // k_cumdecay = attn @ kbe
  for (int tn = tnb; tn < tnb + 4; ++tn) {
    const int bn = tn * 16 + lr;
    v8f c = {};
    #pragma unroll 8
    for (int kk = 0; kk < CHUNK; kk += 4) {
      const int ko = kk + 2 * lh;
      v2f a = *(const v2f*)&attn[am][ko];
      v2f b = *(const v2f*)&vbsT[bn][ko];
      c = wmma4(a, b, c);
    }
    #pragma unroll
    for (int v = 0; v < 8; ++v)
      kcd[base + (size_t)(tm * 16 + v + 8 * lh) * DK + bn] = c[v];
  }
}

// ---------------------------------------------------------------------------
// Sequential inter-chunk scan, one block per head; 128x128 fp32 state in LDS.
// State held TRANSPOSED: ST[v][d] = S[d][v] so B-fragments (and the state
// update's C/D access) are contiguous b64/b128 LDS accesses.
// ---------------------------------------------------------------------------
__global__ __launch_bounds__(256) void scan_kernel(
    const float* __restrict__ qn, const float* __restrict__ kn,
    const float* __restrict__ v_intra, const float* __restrict__ kcd,
    const float* __restrict__ g_h, float* __restrict__ core) {
  __shared__ float ST[DV][LDK];       // state, transposed [v][d]
  __shared__ float nbuf[DV * LDC];    // kcd chunk [s][k] (stride LDK<=LDC*?) / v_new^T [d][s]
  __shared__ float qs[CHUNK][LDK];    // q chunk row-major
  __shared__ float as_[CHUNK][LDC];   // a_intra
  __shared__ float graw[CHUNK], gcum[CHUNK], eg[CHUNK], ew[CHUNK];

  const int h = blockIdx.x;
  const int tid = threadIdx.x, wave = tid >> 5, lane = tid & 31;
  const int lr = lane & 15, lh = lane >> 4;
  const int tm = wave >> 1, tnb = (wave & 1) * 4;   // 4x8 tiling, 4 tiles/wave
  const int am = tm * 16 + lr;

  for (int idx = tid; idx < DV * LDK; idx += 256) (&ST[0][0])[idx] = 0.f;
  __syncthreads();

  for (int n = 0; n < NCHUNK; ++n) {
    const size_t cb = ((size_t)h * LSEQ + n * CHUNK) * DK;
    if (tid < CHUNK) graw[tid] = g_h[h * LSEQ + n * CHUNK + tid];
    __syncthreads();
    if (tid == 0) {
      float s = 0.f;
      for (int i = 0; i < CHUNK; ++i) { s += graw[i]; gcum[i] = s; }
    }
    __syncthreads();
    const float glast = graw[CHUNK - 1];
    if (tid < CHUNK) {
      eg[tid] = expf(graw[tid]);
      ew[tid] = expf(glast - graw[tid]);
    }
    {  // stage kcd chunk row-major [s][k] into nbuf; q chunk into qs
      const int r = tid >> 2, c0 = (tid & 3) * 32;
      for (int j = 0; j < 32; ++j) nbuf[r * LDK + c0 + j] = kcd[cb + (size_t)r * DK + c0 + j];
      for (int j = 0; j < 32; ++j) qs[r][c0 + j] = qn[cb + (size_t)r * DK + c0 + j];
    }
    __syncthreads();

    // step 1: v_new = v_intra - kcd @ S
    v8f av[4];
    #pragma unroll
    for (int t = 0; t < 4; ++t) {
      const int bn = (tnb + t) * 16 + lr;
      v8f c;
      #pragma unroll
      for (int v = 0; v < 8; ++v)
        c[v] = v_intra[cb + (size_t)(tm * 16 + v + 8 * lh) * DV + bn];
      for (int kk = 0; kk < DK; kk += 4) {
        const int ko = kk + 2 * lh;
        v2f a = *(const v2f*)&nbuf[am * LDK + ko];
        a = -a;
        v2f b = *(const v2f*)&ST[bn][ko];
        c = wmma4(a, b, c);
      }
      av[t] = c;
    }
    // step 2: out_acc = (q * exp(g_raw)) @ S
    v8f oacc[4];
    const float ega = eg[am];
    #pragma unroll
    for (int t = 0; t < 4; ++t) {
      const int bn = (tnb + t) * 16 + lr;
      v8f c = {};
      for (int kk = 0; kk < DK; kk += 4) {
        const int ko = kk + 2 * lh;
        v2f a = *(const v2f*)&qs[am][ko];
        a *= ega;
        v2f b = *(const v2f*)&ST[bn][ko];
        c = wmma4(a, b, c);
      }
      oacc[t] = c;
    }
    __syncthreads();  // all reads of staged kcd + old S done

    // publish v_new transposed: nbuf[d * LDC + s]
    #pragma unroll
    for (int t = 0; t < 4; ++t)
      #pragma unroll
      for (int v = 0; v < 8; ++v)
        nbuf[((tnb + t) * 16 + lr) * LDC + tm * 16 + v + 8 * lh] = av[t][v];

    // step 3: a_intra = (q @ k^T) * decay (t <= s), 2 of 16 tiles per wave
    {
      const int t2 = wave * 2;
      for (int tix = t2; tix < t2 + 2; ++tix) {
        const int tm2 = tix >> 2, tn2 = tix & 3;
        const int am2 = tm2 * 16 + lr;
        const float* krow = kn + cb + (size_t)(tn2 * 16 + lr) * DK;
        v8f c = {};
        for (int kk = 0; kk < DK; kk += 4) {
          const int ko = kk + 2 * lh;
          v2f a = *(const v2f*)&qs[am2][ko];
          v2f b = *(const v2f*)(krow + ko);
          c = wmma4(a, b, c);
        }
        #pragma unroll
        for (int v = 0; v < 8; ++v) {
          const int s_ = tm2 * 16 + v + 8 * lh;
          const int t_ = tn2 * 16 + lr;
          as_[s_][t_] = (t_ <= s_) ? c[v] * expf(gcum[s_] - gcum[t_]) : 0.f;
        }
      }
    }
    __syncthreads();

    // step 4: out = attn_inter + a_intra @ v_new  -> core
    #pragma unroll
    for (int t = 0; t < 4; ++t) {
      const int bn = (tnb + t) * 16 + lr;
      v8f c = oacc[t];
      for (int kk = 0; kk < CHUNK; kk += 4) {
        const int ko = kk + 2 * lh;
        v2f a = *(const v2f*)&as_[am][ko];
        v2f b = *(const v2f*)&nbuf[bn * LDC + ko];
        c = wmma4(a, b, c);
      }
      #pragma unroll
      for (int v = 0; v < 8; ++v)
        core[cb + (size_t)(tm * 16 + v + 8 * lh) * DV + bn] = c[v];
    }
    __syncthreads();

    // step 5: S = S*exp(g_last) + kw^T @ v_new (wave w owns d-rows 16w..16w+15)
    {
      const float el = expf(glast);
      const int m0 = wave * 16;
      const float* kcol = kn + cb + (m0 + lr);  // A[m][k] = kn[k][m] * ew[k]
      for (int tn = 0; tn < 8; ++tn) {
        const int bn = tn * 16 + lr;
        v8f c;
        #pragma unroll
        for (int v = 0; v < 8; ++v) c[v] = ST[bn][m0 + v + 8 * lh] * el;
        for (int kk = 0; kk < CHUNK; kk += 4) {
          const int k0 = kk + 2 * lh;
          v2f a, b;
          a.x = kcol[(size_t)k0 * DK] * ew[k0];
          a.y = kcol[(size_t)(k0 + 1) * DK] * ew[k0 + 1];
          b = *(const v2f*)&nbuf[bn * LDC + k0];
          c = wmma4(a, b, c);
        }
        #pragma unroll
        for (int v = 0; v < 8; ++v) ST[bn][m0 + v + 8 * lh] = c[v];
      }
    }
    __syncthreads();
  }
}

// ---------------------------------------------------------------------------
// RMS norm over Dv, * norm_weight, * silu(z); writes (L, 4096) row-major
// ---------------------------------------------------------------------------
__global__ __launch_bounds__(128) void gate_norm_kernel(
    const float* __restrict__ core, const float* __restrict__ qkvz,
    const float* __restrict__ norm_weight, float* __restrict__ gated) {
  const int h = blockIdx.x, l = blockIdx.y, d = threadIdx.x;
  const float x = core[((size_t)h * LSEQ + l) * DV + d];
  __shared__ float red[128];
  red[d] = x * x; __syncthreads();
  for (int s = 64; s > 0; s >>= 1) { if (d < s) red[d] += red[d + s]; __syncthreads(); }
  const float inv = rsqrtf(red[0] * (1.f / DV) + 1e-6f);
  const float normed = x * inv * norm_weight[d];
  const float z = qkvz[(size_t)l * QKVZ_N + (2 * KEY_DIM + VALUE_DIM) + h * DV + d];
  gated[(size_t)l * VALUE_DIM + h * DV + d] = normed * silu_f(z);
}

// ---------------------------------------------------------------------------
extern "C" void kernel_launch(void* const* d_in, const int* in_sizes, int n_in,
                              void* d_out, int out_size, void* d_ws, size_t ws_size,
                              hipStream_t stream) {
  (void)in_sizes; (void)n_in; (void)out_size; (void)ws_size;
  const float* hidden  = (const float*)d_in[0];
  const float* w_qkvz  = (const float*)d_in[1];
  const float* w_ba    = (const float*)d_in[2];
  const float* conv_w  = (const float*)d_in[3];
  const float* conv_b  = (const float*)d_in[4];
  const float* A_log   = (const float*)d_in[5];
  const float* dt_bias = (const float*)d_in[6];
  const float* norm_w  = (const float*)d_in[7];
  const float* w_out   = (const float*)d_in[8];
  float* out = (float*)d_out;

  float* ws = (float*)d_ws;
  size_t off = 0;
  float* qkvz    = ws + off; off += (size_t)LSEQ * QKVZ_N;
  float* ba      = ws + off; off += (size_t)LSEQ * 64;
  float* g_h     = ws + off; off += (size_t)HV * LSEQ;
  float* beta_h  = ws + off; off += (size_t)HV * LSEQ;
  float* qn      = ws + off; off += (size_t)HV * LSEQ * DK;
  float* kn      = ws + off; off += (size_t)HV * LSEQ * DK;
  float* vb      = ws + off; off += (size_t)HV * LSEQ * DV;
  float* v_intra = ws + off; off += (size_t)HV * LSEQ * DV;
  float* kcd     = ws + off; off += (size_t)HV * LSEQ * DK;
  float* core    = ws + off; off += (size_t)HV * LSEQ * DV;
  float* gated   = ws + off; off += (size_t)LSEQ * VALUE_DIM;

  // 1) qkvz = hidden @ w_qkvz ; ba = hidden @ w_ba
  gemm_f32_wmma<<<dim3(LSEQ / 128, QKVZ_N / 64), 256, 0, stream>>>(
      hidden, w_qkvz, qkvz, LSEQ, QKVZ_N, EDIM);
  gemm_f32_wmma<<<dim3(LSEQ / 128, 1), 256, 0, stream>>>(
      hidden, w_ba, ba, LSEQ, 64, EDIM);
  // 2) gates
  gb_kernel<<<(LSEQ * HV + 255) / 256, 256, 0, stream>>>(ba, A_log, dt_bias, g_h, beta_h);
  // 3) conv + silu + l2norm (head-major q/k/v*beta)
  prep_kernel<<<dim3(HV, LSEQ), 128, 0, stream>>>(qkvz, conv_w, conv_b, beta_h, qn, kn, vb);
  // 4) intra-chunk products
  chunk_kernel<<<dim3(NCHUNK, HV), 256, 0, stream>>>(kn, vb, g_h, beta_h, v_intra, kcd);
  // 5) inter-chunk recurrence
  scan_kernel<<<HV, 256, 0, stream>>>(qn, kn, v_intra, kcd, g_h, core);
  // 6) RMS-norm * silu(z) gate
  gate_norm_kernel<<<dim3(HV, LSEQ), 128, 0, stream>>>(core, qkvz, norm_w, gated);
  // 7) out = gated @ w_out
  gemm_f32_wmma<<<dim3(LSEQ / 128, EDIM / 64), 256, 0, stream>>>(
      gated, w_out, out, LSEQ, EDIM, VALUE_DIM);
}